// ProbMPS_38113539784978
// MI455X (gfx1250) — compile-verified
//
#include <hip/hip_runtime.h>
#include <math.h>

// ---------------------------------------------------------------------------
// ProbMPS on MI455X (gfx1250).
//  Kernel 1: log_norm scan: rho <- sum_i A_i rho A_i^T as dense 128^3 fp32
//            GEMMs with V_WMMA_F32_16X16X4_F32. One 512-thread workgroup,
//            16 waves each owning a 32x32 (2x2-tile) output block held in
//            VGPRs across the 16-symbol accumulation. A-matrices double-
//            buffered in LDS via global_load_async_to_lds_b128. LDS rows
//            padded to 132 floats (bank-conflict-free fragment loads);
//            rho read via its symmetry so all fragment loads are b64.
//  Kernel 2: per-batch gathered mat-vec chain; 256 WGs x 128 threads, async
//            double-buffered 64KB matrix staging, skewed LDS dot product.
// ---------------------------------------------------------------------------

#define SEQ   512
#define NI    16
#define DD    128
#define RS    132            // padded LDS row stride (floats): banks = 4*row+k
#define MATP  (DD * RS)      // padded LDS matrix footprint (floats)
#define BATCH 256
#define EPSV  1e-30f

typedef float v2f __attribute__((ext_vector_type(2)));
typedef float v8f __attribute__((ext_vector_type(8)));

// ---- async global -> LDS staging (CDNA5 ASYNCcnt path) --------------------
// Copies a 128x128 f32 matrix (row-major, contiguous in global) into LDS
// with row stride `ldsStride` floats, 16 bytes per lane per instruction.
__device__ __forceinline__ void async_stage_b128(float* dst_lds,
                                                 const float* src,
                                                 int tid, int nthreads,
                                                 int ldsStride)
{
    for (int idx = tid; idx < DD * DD / 4; idx += nthreads) {
        const int row  = idx >> 5;            // 32 float4 chunks per row
        const int col4 = (idx & 31) << 2;
        const unsigned laddr =
            (unsigned)(uintptr_t)(dst_lds + row * ldsStride + col4);
        const unsigned long long gaddr =
            (unsigned long long)(uintptr_t)(src + (size_t)idx * 4);
        asm volatile("global_load_async_to_lds_b128 %0, %1, off"
                     :: "v"(laddr), "v"(gaddr) : "memory");
    }
}

__device__ __forceinline__ void wait_async0()
{
    asm volatile("s_wait_asynccnt 0" ::: "memory");
}

// ---------------------------------------------------------------------------
// Kernel 1: log_norm.  grid = 1 block of 512 threads (16 waves).
// Dynamic LDS: rho[MATP] | Tm[MATP] | As0[MATP] | As1[MATP] | red[64]
// ---------------------------------------------------------------------------
__global__ __launch_bounds__(512)
void probmps_lognorm_kernel(const float* __restrict__ core,
                            const float* __restrict__ edge,
                            float* __restrict__ out_lognorm)
{
    extern __shared__ float sm[];
    float* rho = sm;                 // symmetric density matrix (padded rows)
    float* Tm  = sm + MATP;          // intermediate A @ rho
    float* red = sm + 4 * MATP;      // reduction scratch
    // A buffers live at sm + (2 + cur) * MATP  (offset arithmetic keeps
    // addrspace(3) inference intact -> ds_load, not flat_load)

    const int tid  = threadIdx.x;       // 0..511
    const int wave = tid >> 5;          // 0..15
    const int lane = tid & 31;          // 0..31
    const int br   = wave >> 2;         // 2x2-tile block row (0..3)
    const int bc   = wave & 3;          // 2x2-tile block col (0..3)

    const float* alpha = edge;          // edge_vecs[0]
    const float* omega = edge + DD;     // edge_vecs[1]

    // rho = outer(alpha, alpha)
    for (int idx = tid; idx < DD * DD; idx += 512) {
        int r = idx >> 7, c = idx & (DD - 1);
        rho[r * RS + c] = alpha[r] * alpha[c];
    }

    // Fragment layouts (ISA 7.12.2, wave32):
    //  A 16x4 f32: lanes 0-15: row M=lane, K={0,1}; lanes 16-31: K={2,3}.
    //  B 4x16 f32: lanes 0-15: col N=lane, K={0,1}; lanes 16-31: K={2,3}.
    //  C/D 16x16 : vgpr v, lanes 0-15 -> (M=v,N=lane); 16-31 -> (M=v+8,N=l-16).
    const int halfSel = (lane >> 4) << 1;   // 0 or 2
    const int lan16   = lane & 15;
    const int hi8     = (lane >> 4) << 3;   // 0 or 8

    // preload first A matrix
    async_stage_b128(sm + 2 * MATP, core, tid, 512, RS);
    wait_async0();
    __syncthreads();

    float ls  = 0.0f;                   // identical in every thread
    int   cur = 0;
    v8f accN[2][2] = {};                // Sum_i A_i rho A_i^T, this wave's block

    const int ar0 = (2 * br + 0) * 16 + lan16;   // A-operand rows
    const int ar1 = (2 * br + 1) * 16 + lan16;
    const int nc0 = (2 * bc + 0) * 16 + lan16;   // B-operand columns (as rows)
    const int nc1 = (2 * bc + 1) * 16 + lan16;

    for (int t = 0; t < SEQ; ++t) {
        for (int i = 0; i < NI; ++i) {
            const size_t g = (size_t)t * NI + i;
            // async staging of the next matrix into the other buffer
            if (g + 1 < (size_t)SEQ * NI)
                async_stage_b128(sm + (2 + (cur ^ 1)) * MATP,
                                 core + (g + 1) * DD * DD, tid, 512, RS);
            const float* As = sm + (2 + cur) * MATP;

            // ---- GEMM1: Tm = As @ rho  (rho symmetric: B[k][n]=rho[n][k]) --
            {
                v8f acc[2][2] = {};
                for (int kk = 0; kk < DD; kk += 4) {
                    const int ak = kk + halfSel;
                    v2f a0, a1, b0, b1;
                    a0.x = As[ar0 * RS + ak];   a0.y = As[ar0 * RS + ak + 1];
                    a1.x = As[ar1 * RS + ak];   a1.y = As[ar1 * RS + ak + 1];
                    b0.x = rho[nc0 * RS + ak];  b0.y = rho[nc0 * RS + ak + 1];
                    b1.x = rho[nc1 * RS + ak];  b1.y = rho[nc1 * RS + ak + 1];
                    acc[0][0] = __builtin_amdgcn_wmma_f32_16x16x4_f32(
                        false, a0, false, b0, (short)0, acc[0][0], false, false);
                    acc[0][1] = __builtin_amdgcn_wmma_f32_16x16x4_f32(
                        false, a0, false, b1, (short)0, acc[0][1], false, false);
                    acc[1][0] = __builtin_amdgcn_wmma_f32_16x16x4_f32(
                        false, a1, false, b0, (short)0, acc[1][0], false, false);
                    acc[1][1] = __builtin_amdgcn_wmma_f32_16x16x4_f32(
                        false, a1, false, b1, (short)0, acc[1][1], false, false);
                }
#pragma unroll
                for (int r = 0; r < 2; ++r)
#pragma unroll
                    for (int c = 0; c < 2; ++c) {
                        const int r0 = (2 * br + r) * 16 + hi8;
                        const int c0 = (2 * bc + c) * 16 + lan16;
#pragma unroll
                        for (int v = 0; v < 8; ++v)
                            Tm[(r0 + v) * RS + c0] = acc[r][c][v];
                    }
            }
            __syncthreads();

            // ---- GEMM2: accN += Tm @ As^T   (B[k][n] = As[n][k]) ----
            for (int kk = 0; kk < DD; kk += 4) {
                const int ak = kk + halfSel;
                v2f a0, a1, b0, b1;
                a0.x = Tm[ar0 * RS + ak];   a0.y = Tm[ar0 * RS + ak + 1];
                a1.x = Tm[ar1 * RS + ak];   a1.y = Tm[ar1 * RS + ak + 1];
                b0.x = As[nc0 * RS + ak];   b0.y = As[nc0 * RS + ak + 1];
                b1.x = As[nc1 * RS + ak];   b1.y = As[nc1 * RS + ak + 1];
                accN[0][0] = __builtin_amdgcn_wmma_f32_16x16x4_f32(
                    false, a0, false, b0, (short)0, accN[0][0], false, false);
                accN[0][1] = __builtin_amdgcn_wmma_f32_16x16x4_f32(
                    false, a0, false, b1, (short)0, accN[0][1], false, false);
                accN[1][0] = __builtin_amdgcn_wmma_f32_16x16x4_f32(
                    false, a1, false, b0, (short)0, accN[1][0], false, false);
                accN[1][1] = __builtin_amdgcn_wmma_f32_16x16x4_f32(
                    false, a1, false, b1, (short)0, accN[1][1], false, false);
            }
            // barrier covers: Tm reuse next i, and async buffer completion
            wait_async0();
            __syncthreads();
            cur ^= 1;
        }

        // ---- trace from diagonal-tile registers ----
        float p = 0.0f;
        if (br == bc) {
#pragma unroll
            for (int r = 0; r < 2; ++r) {
                if (lane < 8)        p += accN[r][r][lane];
                else if (lane >= 24) p += accN[r][r][lane - 24];
            }
        }
        for (int off = 16; off > 0; off >>= 1) p += __shfl_xor(p, off, 32);
        if (lane == 0) red[wave] = p;
        __syncthreads();
        float tr = 0.0f;
        for (int w = 0; w < 16; ++w) tr += red[w];
        const float inv = 1.0f / tr;
        ls += logf(tr);
        __syncthreads();

        // ---- rho = accN / tr ; reset accN ----
#pragma unroll
        for (int r = 0; r < 2; ++r)
#pragma unroll
            for (int c = 0; c < 2; ++c) {
                const int r0 = (2 * br + r) * 16 + hi8;
                const int c0 = (2 * bc + c) * 16 + lan16;
#pragma unroll
                for (int v = 0; v < 8; ++v)
                    rho[(r0 + v) * RS + c0] = accN[r][c][v] * inv;
                accN[r][c] = (v8f){};
            }
        __syncthreads();
    }

    // ---- val = omega^T rho omega ----
    float p = 0.0f;
    if (tid < DD) {
        float s = 0.0f;
        for (int c = 0; c < DD; ++c) s += rho[tid * RS + c] * omega[c];
        p = omega[tid] * s;
    }
    for (int off = 16; off > 0; off >>= 1) p += __shfl_xor(p, off, 32);
    if (lane == 0) red[wave] = p;
    __syncthreads();
    if (tid == 0) {
        float val = 0.0f;
        for (int w = 0; w < 16; ++w) val += red[w];
        out_lognorm[0] = logf(val) + ls;
    }
}

// ---------------------------------------------------------------------------
// Kernel 2: batch chains.  grid = 256 blocks (one batch each) x 128 threads.
// Dynamic LDS: As0[128*128] | As1[128*128] | hs[128] | red[8]
// ---------------------------------------------------------------------------
__global__ __launch_bounds__(128)
void probmps_batch_kernel(const int*   __restrict__ input,
                          const float* __restrict__ core,
                          const float* __restrict__ edge,
                          const float* __restrict__ lognorm_ws,
                          float* __restrict__ out)
{
    extern __shared__ float sm[];
    float* hs  = sm + 2 * DD * DD;
    float* red = sm + 2 * DD * DD + DD;
    // matrix buffers at sm + (t&1)*DD*DD (offset arithmetic -> ds path)

    const int tid  = threadIdx.x;    // 0..127
    const int lane = tid & 31;
    const int wave = tid >> 5;
    const int b    = blockIdx.x;

    const float* alpha = edge;
    const float* omega = edge + DD;

    hs[tid] = alpha[tid];

    // preload t=0 gathered matrix
    {
        const int x0 = input[b * SEQ];
        async_stage_b128(sm, core + (size_t)x0 * DD * DD, tid, 128, DD);
    }
    wait_async0();
    __syncthreads();

    float ls = 0.0f;   // identical in every thread

    for (int t = 0; t < SEQ; ++t) {
        const float* As = sm + (t & 1) * DD * DD;

        // async staging of next step's gathered matrix into the other buffer
        if (t + 1 < SEQ) {
            const int xn = input[b * SEQ + t + 1];
            async_stage_b128(sm + ((t + 1) & 1) * DD * DD,
                             core + ((size_t)(t + 1) * NI + xn) * DD * DD,
                             tid, 128, DD);
        }

        // ---- h_new[r] = dot(A[r,:], h), diagonally skewed (bank-conflict
        //      free: bank = (r + j) mod 64 distinct across a wave) ----
        float s = 0.0f;
        const int base = tid * DD;
        for (int j = 0; j < DD; ++j) {
            const int c = (tid + j) & (DD - 1);
            s += As[base + c] * hs[c];
        }

        // ---- L2 norm across 128 threads ----
        float p = s * s;
        for (int off = 16; off > 0; off >>= 1) p += __shfl_xor(p, off, 32);
        if (lane == 0) red[wave] = p;
        __syncthreads();
        const float nrm2  = red[0] + red[1] + red[2] + red[3];
        const float scale = fmaxf(sqrtf(nrm2), EPSV);
        ls += logf(scale);
        hs[tid] = s / scale;

        // buffer for t+1 ready + hs visible before next iteration
        wait_async0();
        __syncthreads();
    }

    // ---- psi = h . omega ; out = 2*(log(max(|psi|,eps)) + ls) - log_norm ----
    float p = hs[tid] * omega[tid];
    for (int off = 16; off > 0; off >>= 1) p += __shfl_xor(p, off, 32);
    if (lane == 0) red[wave] = p;
    __syncthreads();
    if (tid == 0) {
        const float psi = red[0] + red[1] + red[2] + red[3];
        const float lu  = logf(fmaxf(fabsf(psi), EPSV)) + ls;
        out[b] = 2.0f * lu - lognorm_ws[0];
    }
}

// ---------------------------------------------------------------------------
extern "C" void kernel_launch(void* const* d_in, const int* in_sizes, int n_in,
                              void* d_out, int out_size, void* d_ws, size_t ws_size,
                              hipStream_t stream)
{
    const int*   input = (const int*)  d_in[0];   // (256, 512) int32
    const float* core  = (const float*)d_in[1];   // (512, 16, 128, 128) f32
    const float* edge  = (const float*)d_in[2];   // (2, 128) f32
    float* out = (float*)d_out;                   // (256,) f32
    float* ws  = (float*)d_ws;                    // ws[0] = log_norm

    const size_t smem1 = (size_t)(4 * MATP + 64) * sizeof(float);         // ~264.3 KB
    const size_t smem2 = (size_t)(2 * DD * DD + DD + 8) * sizeof(float);  // ~128.5 KB

    (void)hipFuncSetAttribute((const void*)probmps_lognorm_kernel,
                              hipFuncAttributeMaxDynamicSharedMemorySize,
                              (int)smem1);
    (void)hipFuncSetAttribute((const void*)probmps_batch_kernel,
                              hipFuncAttributeMaxDynamicSharedMemorySize,
                              (int)smem2);

    probmps_lognorm_kernel<<<1, 512, smem1, stream>>>(core, edge, ws);
    probmps_batch_kernel<<<BATCH, 128, smem2, stream>>>(input, core, edge, ws, out);
}